// IFStransforms_20529943675508
// MI455X (gfx1250) — compile-verified
//
#include <hip/hip_runtime.h>
#include <math.h>

typedef __attribute__((ext_vector_type(2))) float v2f;
typedef __attribute__((ext_vector_type(8))) float v8f;

#define N_AFF 4
#define DEPTH 6
#define NT    4096   /* 4^DEPTH */
#define BATCH 8192

// ---------------------------------------------------------------------------
// Phase 1: compose all NT = 4^DEPTH products of the inverse affines.
// w[t*8 + {0..7}] = { m00, m01, m02, scale_prod, m10, m11, m12, 0 }
// Tm[t] = inv(A[n1]) @ ... @ inv(A[n6]), t = ((((n1*4+n2)*4+n3)*4+n4)*4+n5)*4+n6
// scale_prod = prod affine_scales[ni]  (reference divides by prod(1/s)).
// ---------------------------------------------------------------------------
__global__ __launch_bounds__(256) void ifs_compose_kernel(
    const float* __restrict__ affine_scales,   // (4,)
    const float* __restrict__ affine_mats,     // (4,3,3) row-major
    float* __restrict__ w)                     // (NT,8)
{
  float inv[N_AFF][3][3];
  float sc[N_AFF];
#pragma unroll
  for (int n = 0; n < N_AFF; ++n) {
    const float* m = affine_mats + n * 9;
    float a = m[0], b = m[1], c = m[2];
    float d = m[3], e = m[4], f = m[5];
    float g = m[6], h = m[7], i = m[8];
    float A =  (e * i - f * h);
    float B = -(d * i - f * g);
    float C =  (d * h - e * g);
    float det = a * A + b * B + c * C;
    float r = 1.0f / det;
    inv[n][0][0] =  A * r;
    inv[n][0][1] = -(b * i - c * h) * r;
    inv[n][0][2] =  (b * f - c * e) * r;
    inv[n][1][0] =  B * r;
    inv[n][1][1] =  (a * i - c * g) * r;
    inv[n][1][2] = -(a * f - c * d) * r;
    inv[n][2][0] =  C * r;
    inv[n][2][1] = -(a * h - b * g) * r;
    inv[n][2][2] =  (a * e - b * d) * r;
    sc[n] = affine_scales[n];
  }

  for (int t = (int)threadIdx.x; t < NT; t += (int)blockDim.x) {
    int dig[DEPTH];
#pragma unroll
    for (int j = 0; j < DEPTH; ++j) dig[j] = (t >> (2 * (DEPTH - 1 - j))) & 3;

    float M[3][3];
#pragma unroll
    for (int r = 0; r < 3; ++r)
#pragma unroll
      for (int c = 0; c < 3; ++c) M[r][c] = inv[dig[0]][r][c];
    float fprod = sc[dig[0]];

#pragma unroll
    for (int j = 1; j < DEPTH; ++j) {
      int n = dig[j];
      float R[3][3];
#pragma unroll
      for (int r = 0; r < 3; ++r)
#pragma unroll
        for (int c = 0; c < 3; ++c)
          R[r][c] = M[r][0] * inv[n][0][c] + M[r][1] * inv[n][1][c] +
                    M[r][2] * inv[n][2][c];
#pragma unroll
      for (int r = 0; r < 3; ++r)
#pragma unroll
        for (int c = 0; c < 3; ++c) M[r][c] = R[r][c];
      fprod *= sc[n];
    }

    float* o = w + (size_t)t * 8;
    o[0] = M[0][0]; o[1] = M[0][1]; o[2] = M[0][2]; o[3] = fprod;
    o[4] = M[1][0]; o[5] = M[1][1]; o[6] = M[1][2]; o[7] = 0.0f;
  }
}

// ---------------------------------------------------------------------------
// Phase 2: wave32 owns 16 queries x one transform group (blockIdx.y).
// A (16x4 f32): lanes 0-15 = {qx,qy} (K=0,1); lanes 16-31 = {1,0} (K=2,3).
// Per 16-transform tile: two V_WMMA_F32_16X16X4_F32 -> x/y coords of
// 16 queries x 16 transforms. Lane n owns column tb+n, 8 query rows; 4 circle
// SDFs each (raw v_sqrt_f32), min-accumulate; butterfly min per half-wave.
// Transform-table loads are software-pipelined one tile ahead. Since fsc>0,
// (sqrt(q)-r)*fsc is computed as fma(sqrt(q), fsc, -r*fsc) with per-tile
// prescaled radii.
// dest[blockIdx.y * BATCH + q] receives the per-group partial min.
// ---------------------------------------------------------------------------
__global__ __launch_bounds__(256) void ifs_eval_kernel(
    const float* __restrict__ query,   // (BATCH,2)
    const float* __restrict__ w,       // (NT,8)
    const float* __restrict__ sdf_c,   // (4,2)
    const float* __restrict__ sdf_r,   // (4,)
    float* __restrict__ dest,          // (groups, BATCH)
    int tcnt)                          // transforms per group
{
  const int lane  = (int)threadIdx.x & 31;
  const int wid   = (int)threadIdx.x >> 5;
  const int qbase = ((int)blockIdx.x * 8 + wid) * 16;
  const int tbeg  = (int)blockIdx.y * tcnt;

  // A operand (queries, homogeneous pad).
  v2f a;
  if (lane < 16) {
    a.x = query[(qbase + lane) * 2 + 0];
    a.y = query[(qbase + lane) * 2 + 1];
  } else {
    a.x = 1.0f;
    a.y = 0.0f;
  }

  float cx[4], cy[4], rr[4];
#pragma unroll
  for (int s = 0; s < 4; ++s) {
    cx[s] = sdf_c[s * 2 + 0];
    cy[s] = sdf_c[s * 2 + 1];
    rr[s] = sdf_r[s];
  }

  float acc[8];
#pragma unroll
  for (int r = 0; r < 8; ++r) acc[r] = 3.402823466e+38f;

  // Software-pipelined transform-table reads (one 16-transform tile ahead).
  const float* wl = w + (size_t)(tbeg + (lane & 15)) * 8;
  float4 w0 = *(const float4*)(wl);      // m00 m01 m02 fprod
  float4 w1 = *(const float4*)(wl + 4);  // m10 m11 m12 0

  for (int i = 0; i < tcnt; i += 16) {
    const float* wn = wl + 16 * 8;
    float4 p0 = {}, p1 = {};
    const bool more = (i + 16) < tcnt;
    if (more) {
      p0 = *(const float4*)(wn);
      p1 = *(const float4*)(wn + 4);
    }

    v2f bx, by;
    if (lane < 16) {               // rows K=0,1 of B
      bx.x = w0.x; bx.y = w0.y;    // B_x[0][n]=m00, B_x[1][n]=m01
      by.x = w1.x; by.y = w1.y;    // B_y[0][n]=m10, B_y[1][n]=m11
    } else {                       // rows K=2,3 of B
      bx.x = w0.z; bx.y = 0.0f;    // B_x[2][n]=m02, B_x[3][n]=0
      by.x = w1.z; by.y = 0.0f;
    }
    const float fsc = w0.w;        // same t for lane n and n+16

    // Prescale radii once per tile: d*fsc = sqrt(q)*fsc - rr*fsc.
    float rf[4];
#pragma unroll
    for (int s = 0; s < 4; ++s) rf[s] = rr[s] * fsc;

    v8f zero = {};
    v8f X = __builtin_amdgcn_wmma_f32_16x16x4_f32(
        false, a, false, bx, (short)0, zero, false, false);
    v8f Y = __builtin_amdgcn_wmma_f32_16x16x4_f32(
        false, a, false, by, (short)0, zero, false, false);

#pragma unroll
    for (int r = 0; r < 8; ++r) {
      float x = X[r];
      float y = Y[r];
      float d[4];
#pragma unroll
      for (int s = 0; s < 4; ++s) {
        float dx = x - cx[s];
        float dy = y - cy[s];
        // raw v_sqrt_f32 (~1 ulp), then single fma folds the scale & radius
        float l = __builtin_amdgcn_sqrtf(fmaf(dx, dx, dy * dy));
        d[s] = fmaf(l, fsc, -rf[s]);
      }
      // tree min (compiler can form v_min3_num_f32)
      acc[r] = fminf(acc[r], fminf(fminf(d[0], d[1]), fminf(d[2], d[3])));
    }

    wl = wn;
    w0 = p0;
    w1 = p1;
  }

  // Min-reduce across the 16 transform columns inside each half-wave.
#pragma unroll
  for (int off = 1; off < 16; off <<= 1) {
#pragma unroll
    for (int r = 0; r < 8; ++r)
      acc[r] = fminf(acc[r], __shfl_xor(acc[r], off, 32));
  }

  // C/D layout: lanes 0-15 hold rows M=0..7 (VGPR r -> row r),
  // lanes 16-31 hold rows M=8..15.
  float* o = dest + (size_t)blockIdx.y * BATCH + qbase;
  if (lane == 0) {
#pragma unroll
    for (int r = 0; r < 8; ++r) o[r] = acc[r];
  }
  if (lane == 16) {
#pragma unroll
    for (int r = 0; r < 8; ++r) o[8 + r] = acc[r];
  }
}

// ---------------------------------------------------------------------------
// Phase 3: deterministic min over the per-group partials.
// ---------------------------------------------------------------------------
__global__ __launch_bounds__(256) void ifs_reduce_kernel(
    const float* __restrict__ partial,  // (groups, BATCH)
    float* __restrict__ out,            // (BATCH,)
    int groups)
{
  int q = (int)blockIdx.x * (int)blockDim.x + (int)threadIdx.x;
  if (q >= BATCH) return;
  float m = partial[q];
  for (int g = 1; g < groups; ++g)
    m = fminf(m, partial[(size_t)g * BATCH + q]);
  out[q] = m;
}

// ---------------------------------------------------------------------------
extern "C" void kernel_launch(void* const* d_in, const int* in_sizes, int n_in,
                              void* d_out, int out_size, void* d_ws,
                              size_t ws_size, hipStream_t stream) {
  const float* query   = (const float*)d_in[0];  // (8192,2)
  const float* scales  = (const float*)d_in[1];  // (4,)
  const float* mats    = (const float*)d_in[2];  // (4,3,3)
  const float* centers = (const float*)d_in[3];  // (4,2)
  const float* radii   = (const float*)d_in[4];  // (4,)
  // d_in[5] = depth (==6, compiled in as DEPTH)
  float* out = (float*)d_out;
  float* w   = (float*)d_ws;                     // table: NT*8*4 = 128 KB

  const size_t table_bytes = (size_t)NT * 8 * sizeof(float);

  // Split transforms across groups for occupancy if scratch allows it.
  int groups = 8;
  if (ws_size < table_bytes + (size_t)BATCH * groups * sizeof(float))
    groups = 1;

  hipLaunchKernelGGL(ifs_compose_kernel, dim3(1), dim3(256), 0, stream,
                     scales, mats, w);

  if (groups == 1) {
    hipLaunchKernelGGL(ifs_eval_kernel, dim3(BATCH / 128, 1), dim3(256), 0,
                       stream, query, w, centers, radii, out, NT);
  } else {
    float* partial = w + (size_t)NT * 8;         // (groups, BATCH)
    hipLaunchKernelGGL(ifs_eval_kernel, dim3(BATCH / 128, groups), dim3(256),
                       0, stream, query, w, centers, radii, partial,
                       NT / groups);
    hipLaunchKernelGGL(ifs_reduce_kernel, dim3((BATCH + 255) / 256), dim3(256),
                       0, stream, partial, out, groups);
  }
}